// SimpleLSTM_52080773431937
// MI455X (gfx1250) — compile-verified
//
#include <hip/hip_runtime.h>
#include <math.h>

// ---------------------------------------------------------------------------
// Persistent-WMMA LSTM for MI455X (gfx1250).
//   B=64, T=1024, I=256, H=512, fp32.
//   - xt_kernel: transpose inputs -> xT[t][i/2][b][i&1] (coalesced A-frag loads)
//   - lstm_kernel (persistent, 128 WGs x 128 thr):
//       * each WG owns 16 columns of the 2048-wide [i|f|o|c] gate matrix
//       * its 768x16 fused weight slice (W_h on K=0..511, W_x on K=512..767)
//         lives in LDS for all 1024 steps (49 KB, K-pair packed, conflict-free)
//       * each wave computes one 16x16 fp32 tile via V_WMMA_F32_16X16X4_F32,
//         192 WMMAs per tile per step (K=768)
//       * grid barrier (atomic + s_sleep), then all 16384 threads do the
//         elementwise c/h update and write h back transposed + output
// Workspace: hT 128KB | c 128KB | gates 512KB | xT 64MB | counter  (~68 MB)
// ---------------------------------------------------------------------------

typedef float v2f __attribute__((ext_vector_type(2)));
typedef float v8f __attribute__((ext_vector_type(8)));

#define SEQ_T   1024
#define DIM_I   256
#define DIM_H   512
#define DIM_B   64
#define NWG     128
#define WGTHR   128

// ws float offsets
#define OFF_HT   0            // 512*64   packed [hc>>1][m][hc&1]
#define OFF_C    32768        // 64*512   [m][hc]
#define OFF_G    65536        // 4*64*512 [g][m][hc]
#define OFF_XT   196608       // 1024*256*64 packed [t][i>>1][b][i&1]
#define OFF_CNT  16973824     // unsigned counter

__global__ void init_kernel(float* __restrict__ ws, unsigned* __restrict__ cnt) {
    int e = blockIdx.x * 256 + threadIdx.x;
    if (e < 65536) ws[e] = 0.0f;           // zero h and c state
    if (e == 0) *cnt = 0u;
}

__global__ void xt_kernel(const float* __restrict__ in, float* __restrict__ xT) {
    int e = blockIdx.x * 256 + threadIdx.x;      // e < 64*1024*256, read-coalesced
    int i = e & 255;
    int t = (e >> 8) & 1023;
    int b = e >> 18;
    xT[t * 16384 + ((i >> 1) << 7) + (b << 1) + (i & 1)] = in[e];
}

__device__ __forceinline__ void grid_barrier(unsigned* cnt, unsigned target) {
    __syncthreads();
    if (threadIdx.x == 0) {
        __threadfence();
        __hip_atomic_fetch_add(cnt, 1u, __ATOMIC_ACQ_REL, __HIP_MEMORY_SCOPE_AGENT);
        while (__hip_atomic_load(cnt, __ATOMIC_ACQUIRE, __HIP_MEMORY_SCOPE_AGENT) < target)
            __builtin_amdgcn_s_sleep(1);
    }
    __syncthreads();
}

__device__ __forceinline__ float fast_sigmoid(float x) {
    return 1.0f / (1.0f + __expf(-x));
}
__device__ __forceinline__ float fast_tanh(float x) {
    return 1.0f - 2.0f / (__expf(2.0f * x) + 1.0f);
}

__global__ __launch_bounds__(WGTHR)
void lstm_kernel(const float* __restrict__ xT,
                 const float* __restrict__ Wh0, const float* __restrict__ Wh1,
                 const float* __restrict__ Wh2, const float* __restrict__ Wh3,
                 const float* __restrict__ Wx0, const float* __restrict__ Wx1,
                 const float* __restrict__ Wx2, const float* __restrict__ Wx3,
                 const float* __restrict__ B0,  const float* __restrict__ B1,
                 const float* __restrict__ B2,  const float* __restrict__ B3,
                 float* __restrict__ hT, float* __restrict__ cbuf,
                 float* __restrict__ gbuf, float* __restrict__ out,
                 unsigned* __restrict__ cnt)
{
    const int wg   = blockIdx.x;        // 0..127
    const int tid  = threadIdx.x;       // 0..127
    const int lane = tid & 31;
    const int wave = tid >> 5;          // 0..3 -> M-tile

    const int g      = wg >> 5;         // gate 0..3 (i,f,o,c)
    const int hcbase = (wg & 31) << 4;  // 16-col slice inside gate

    const float* Wh = (g == 0) ? Wh0 : (g == 1) ? Wh1 : (g == 2) ? Wh2 : Wh3;
    const float* Wx = (g == 0) ? Wx0 : (g == 1) ? Wx1 : (g == 2) ? Wx2 : Wx3;
    const float* Bv = (g == 0) ? B0  : (g == 1) ? B1  : (g == 2) ? B2  : B3;

    // Fused weight slice, K-pair packed: lw[(k>>1)*32 + col*2 + (k&1)].
    // Row-pair stride 32 floats => lanes 0-15 (K pair q) and 16-31 (pair q+1)
    // hit disjoint 32-bank halves: conflict-free ds_load_b64.
    __shared__ float lw[384 * 32];      // 49152 B
    for (int idx = tid; idx < 768 * 16; idx += WGTHR) {
        int k = idx >> 4, cl = idx & 15;
        float v = (k < DIM_H) ? Wh[k * DIM_H + hcbase + cl]
                              : Wx[(k - DIM_H) * DIM_H + hcbase + cl];
        lw[(k >> 1) * 32 + cl * 2 + (k & 1)] = v;
    }
    __syncthreads();

    // WMMA fragment coordinates (ISA 7.12.2, f32 16x16x4):
    //  A: lane holds A[lane&15][k0 + j],  k0 = 2*(lane>>4)
    //  B: lane holds B[k0 + j][lane&15]
    //  C/D vgpr r: lane holds C[r + 8*(lane>>4)][lane&15]
    const int mloc  = lane & 15;
    const int khalf = lane >> 4;
    const int k0    = khalf << 1;
    const int r2    = ((wave << 4) + mloc) << 1;   // A row *2 (pair-packed)
    const int nl2   = mloc << 1;                   // B col *2
    const int hc    = hcbase + mloc;
    const float bb  = Bv[hc];
    const int  goff = g << 15;                     // g * 32768

    unsigned epoch = 0;
    for (int t = 0; t < SEQ_T; ++t) {
        v8f acc = {bb, bb, bb, bb, bb, bb, bb, bb};

        // ---- h @ W_h  (K = 0..511), A from transposed h state ----
        #pragma unroll 4
        for (int k = 0; k < DIM_H; k += 4) {
            const int kk = k + k0;                                  // even
            v2f a = *(const v2f*)(hT + ((kk >> 1) << 7) + r2);
            v2f b = *(const v2f*)(lw + (kk >> 1) * 32 + nl2);
            acc = __builtin_amdgcn_wmma_f32_16x16x4_f32(
                      false, a, false, b, (short)0, acc, false, false);
        }
        // ---- x_t @ W_x  (K = 512..767), A from transposed inputs ----
        const float* xTt = xT + t * 16384;
        #pragma unroll 4
        for (int k = 0; k < DIM_I; k += 4) {
            const int kk = k + k0;                                  // even
            v2f a = *(const v2f*)(xTt + ((kk >> 1) << 7) + r2);
            v2f b = *(const v2f*)(lw + ((kk + DIM_H) >> 1) * 32 + nl2);
            acc = __builtin_amdgcn_wmma_f32_16x16x4_f32(
                      false, a, false, b, (short)0, acc, false, false);
        }

        // store raw pre-activations: gbuf[g][m][hc]
        #pragma unroll
        for (int r = 0; r < 8; ++r) {
            int mrow = (wave << 4) + r + (khalf << 3);
            gbuf[goff + (mrow << 9) + hc] = acc[r];
        }

        // prefetch next timestep's x slab into caches while we sync
        if (t + 1 < SEQ_T) {
            const float* nx = xT + (t + 1) * 16384;
            #pragma unroll
            for (int p = 0; p < 4; ++p)
                __builtin_prefetch(nx + ((tid + p * 128) << 5), 0, 0);
        }

        ++epoch;
        grid_barrier(cnt, epoch * (unsigned)NWG);

        // ---- elementwise LSTM cell update: 32768 elems over 16384 threads ----
        const int gid = (wg << 7) + tid;
        #pragma unroll
        for (int e = gid; e < DIM_B * DIM_H; e += NWG * WGTHR) {
            int m = e >> 9, hcol = e & 511;
            float It = fast_sigmoid(gbuf[e]);
            float Ft = fast_sigmoid(gbuf[32768 + e]);
            float Ot = fast_sigmoid(gbuf[65536 + e]);
            float Ct = fast_tanh   (gbuf[98304 + e]);
            float c  = Ft * cbuf[e] + It * Ct;
            cbuf[e]  = c;
            float hn = Ot * fast_tanh(c);
            hT[((hcol >> 1) << 7) + (m << 1) + (hcol & 1)] = hn;   // h state (packed)
            out[(m << 19) + (t << 9) + hcol] = hn;                  // (B,T,H)
        }

        ++epoch;
        grid_barrier(cnt, epoch * (unsigned)NWG);
    }
}

extern "C" void kernel_launch(void* const* d_in, const int* in_sizes, int n_in,
                              void* d_out, int out_size, void* d_ws, size_t ws_size,
                              hipStream_t stream) {
    (void)in_sizes; (void)n_in; (void)out_size;
    if (ws_size < (size_t)(OFF_CNT) * 4 + 16) return;   // need ~68 MB scratch

    const float* inputs = (const float*)d_in[0];
    const float* Wh[4]  = {(const float*)d_in[1], (const float*)d_in[4],
                           (const float*)d_in[7], (const float*)d_in[10]};
    const float* Wx[4]  = {(const float*)d_in[2], (const float*)d_in[5],
                           (const float*)d_in[8], (const float*)d_in[11]};
    const float* Bb[4]  = {(const float*)d_in[3], (const float*)d_in[6],
                           (const float*)d_in[9], (const float*)d_in[12]};

    float*    ws   = (float*)d_ws;
    float*    hT   = ws + OFF_HT;
    float*    cbuf = ws + OFF_C;
    float*    gbuf = ws + OFF_G;
    float*    xT   = ws + OFF_XT;
    unsigned* cnt  = (unsigned*)(ws + OFF_CNT);
    float*    out  = (float*)d_out;

    init_kernel<<<257, 256, 0, stream>>>(ws, cnt);
    xt_kernel<<<65536, 256, 0, stream>>>(inputs, xT);
    lstm_kernel<<<NWG, WGTHR, 0, stream>>>(xT,
        Wh[0], Wh[1], Wh[2], Wh[3],
        Wx[0], Wx[1], Wx[2], Wx[3],
        Bb[0], Bb[1], Bb[2], Bb[3],
        hT, cbuf, gbuf, out, cnt);
}